// PoseLossV2_13159779795087
// MI455X (gfx1250) — compile-verified
//
#include <hip/hip_runtime.h>
#include <math.h>

typedef __attribute__((ext_vector_type(2))) float v2f;
typedef __attribute__((ext_vector_type(4))) float v4f;
typedef __attribute__((ext_vector_type(8))) float v8f;

#define NB   32
#define NK   17
#define HW   16384          // 128*128
#define WIDTH 128
#define BK   (NB * NK)      // 544
#define PAF_ELEMS (32 * 38 * 128 * 128)   // 19,922,944
#define PAF_N4    (PAF_ELEMS / 4)         // 4,980,736
#define PAF_BLOCKS 1024
#define PAF_ITERS  (PAF_N4 / (PAF_BLOCKS * 256))  // 19 exactly

// COCO skeleton edges
__constant__ int c_ea[19] = {15,13,16,14,11, 5, 6, 5, 5, 6, 7, 8, 1, 0, 0, 1, 2, 3, 4};
__constant__ int c_eb[19] = {13,11,14,12,12,11,12, 6, 7, 8, 9,10, 2, 1, 2, 3, 4, 5, 6};

// Wave32 sum reduction via V_WMMA_F32_16X16X4_F32.
// A layout (16x4 f32, ISA-documented): lanes 0-15 VGPR0 = K=0 (M=lane),
// lanes 16-31 VGPR0 = K=2 (M=lane-16); VGPR1 = K=1 / K=3 (set to 0 here).
// B = all-ones (value-uniform, so its lane layout is irrelevant), C = 0.
// D[m][n] = s[m] + s[m+16]. Summing the 8 D VGPRs gives rows 0-7 on lanes
// 0-15 and rows 8-15 on lanes 16-31; one shfl_xor(16) completes the sum.
// Requires EXEC == all ones (call sites are fully convergent).
__device__ __forceinline__ float wave_sum32(float s) {
    v2f a; a[0] = s;    a[1] = 0.0f;
    v2f b; b[0] = 1.0f; b[1] = 1.0f;
    v8f c = {};
    v8f d = __builtin_amdgcn_wmma_f32_16x16x4_f32(
        /*neg_a=*/false, a, /*neg_b=*/false, b,
        /*c_mod=*/(short)0, c, /*reuse_a=*/false, /*reuse_b=*/false);
    float p = d[0] + d[1] + d[2] + d[3] + d[4] + d[5] + d[6] + d[7];
    p += __shfl_xor(p, 16, 32);
    return p;   // full 32-lane sum, broadcast to all lanes
}

// Kernel 1: one block per (b,k) heatmap tile. Fused MSE-sum + flat argmax.
__global__ void __launch_bounds__(256)
heatmap_kernel(const float* __restrict__ pred, const float* __restrict__ gt,
               float* __restrict__ per_kp, int* __restrict__ amax) {
    const int bk = blockIdx.x;
    const int t  = threadIdx.x;
    const v4f* p4 = (const v4f*)(pred + (size_t)bk * HW);
    const v4f* g4 = (const v4f*)(gt   + (size_t)bk * HW);

    float sum = 0.0f;
    float mv  = -INFINITY;
    int   mi  = 0x7fffffff;

    #pragma unroll
    for (int i = 0; i < 16; ++i) {
        const int vidx = i * 256 + t;
        v4f p = __builtin_nontemporal_load(&p4[vidx]);   // streamed once: TH=NT
        v4f g = __builtin_nontemporal_load(&g4[vidx]);
        float d0 = p[0] - g[0], d1 = p[1] - g[1];
        float d2 = p[2] - g[2], d3 = p[3] - g[3];
        sum += d0 * d0 + d1 * d1 + d2 * d2 + d3 * d3;
        const int base = vidx * 4;
        // strict '>' keeps the first occurrence (indices grow per thread)
        if (p[0] > mv) { mv = p[0]; mi = base;     }
        if (p[1] > mv) { mv = p[1]; mi = base + 1; }
        if (p[2] > mv) { mv = p[2]; mi = base + 2; }
        if (p[3] > mv) { mv = p[3]; mi = base + 3; }
    }

    // wave-level reductions (fully convergent here)
    float wsum = wave_sum32(sum);
    #pragma unroll
    for (int off = 16; off > 0; off >>= 1) {
        float ov = __shfl_xor(mv, off, 32);
        int   oi = __shfl_xor(mi, off, 32);
        if (ov > mv || (ov == mv && oi < mi)) { mv = ov; mi = oi; }
    }

    __shared__ float s_sum[8];
    __shared__ float s_max[8];
    __shared__ int   s_idx[8];
    const int wave = t >> 5, lane = t & 31;
    if (lane == 0) { s_sum[wave] = wsum; s_max[wave] = mv; s_idx[wave] = mi; }
    __syncthreads();

    if (t == 0) {
        float s = 0.0f, bm = -INFINITY; int bi = 0x7fffffff;
        #pragma unroll
        for (int w = 0; w < 8; ++w) {
            s += s_sum[w];
            if (s_max[w] > bm || (s_max[w] == bm && s_idx[w] < bi)) {
                bm = s_max[w]; bi = s_idx[w];
            }
        }
        per_kp[bk] = s * (1.0f / (float)HW);
        amax[bk]   = bi;
    }
}

// Kernel 2: PAF MSE streaming reduction; fixed per-block partials (no atomics).
__global__ void __launch_bounds__(256)
paf_kernel(const float* __restrict__ a, const float* __restrict__ b,
           float* __restrict__ partials) {
    const v4f* a4 = (const v4f*)a;
    const v4f* b4 = (const v4f*)b;
    const int t   = threadIdx.x;
    const int gid = blockIdx.x * 256 + t;

    float sum = 0.0f;
    #pragma unroll
    for (int i = 0; i < PAF_ITERS; ++i) {
        const int idx = i * (PAF_BLOCKS * 256) + gid;
        v4f x = __builtin_nontemporal_load(&a4[idx]);
        v4f y = __builtin_nontemporal_load(&b4[idx]);
        float d0 = x[0] - y[0], d1 = x[1] - y[1];
        float d2 = x[2] - y[2], d3 = x[3] - y[3];
        sum += d0 * d0 + d1 * d1 + d2 * d2 + d3 * d3;
    }

    float wsum = wave_sum32(sum);
    __shared__ float s_sum[8];
    const int wave = t >> 5, lane = t & 31;
    if (lane == 0) s_sum[wave] = wsum;
    __syncthreads();
    if (t == 0) {
        float s = 0.0f;
        #pragma unroll
        for (int w = 0; w < 8; ++w) s += s_sum[w];
        partials[blockIdx.x] = s;
    }
}

// Kernel 3: single-block finalize. Top-8 OHKM, struct loss, PAF partial sum.
__global__ void __launch_bounds__(256)
finalize_kernel(const float* __restrict__ per_kp, const int* __restrict__ amax,
                const float* __restrict__ partials, const int* __restrict__ coords,
                float* __restrict__ out) {
    const int t = threadIdx.x;
    __shared__ float s_hm[NB];
    __shared__ float s_st[NB];
    __shared__ float s_paf[256];

    // deterministic fixed-assignment partial gather (1024 partials -> 256)
    float ps = partials[t] + partials[t + 256] + partials[t + 512] + partials[t + 768];
    s_paf[t] = ps;

    if (t < NB) {
        const int bb = t;
        // OHKM: mean of top-8 per-keypoint losses
        float v[NK];
        #pragma unroll
        for (int k = 0; k < NK; ++k) v[k] = per_kp[bb * NK + k];
        float acc = 0.0f;
        #pragma unroll
        for (int i = 0; i < 8; ++i) {
            int mj = 0; float mvv = v[0];
            #pragma unroll
            for (int k = 1; k < NK; ++k) if (v[k] > mvv) { mvv = v[k]; mj = k; }
            acc += mvv;
            v[mj] = -INFINITY;
        }
        s_hm[bb] = acc * (1.0f / 8.0f);

        // structural loss
        float px[NK], py[NK], gx[NK], gy[NK];
        #pragma unroll
        for (int k = 0; k < NK; ++k) {
            const int idx = amax[bb * NK + k];
            px[k] = (float)(idx & (WIDTH - 1));
            py[k] = (float)(idx >> 7);
            gx[k] = (float)coords[(bb * NK + k) * 2 + 0];
            gy[k] = (float)coords[(bb * NK + k) * 2 + 1];
        }
        float sacc = 0.0f;
        #pragma unroll
        for (int e = 0; e < 19; ++e) {
            const int ea = c_ea[e], eb = c_eb[e];
            float dpx = px[ea] - px[eb], dpy = py[ea] - py[eb];
            float dgx = gx[ea] - gx[eb], dgy = gy[ea] - gy[eb];
            float pl = sqrtf(dpx * dpx + dpy * dpy);
            float gl = sqrtf(dgx * dgx + dgy * dgy);
            float d  = pl - gl;
            sacc += d * d;
        }
        s_st[bb] = sacc / (19.0f + 1e-6f);
    }
    __syncthreads();

    // deterministic tree reduction of PAF partials
    for (int off = 128; off > 0; off >>= 1) {
        if (t < off) s_paf[t] += s_paf[t + off];
        __syncthreads();
    }

    if (t == 0) {
        float hm = 0.0f, st = 0.0f;
        #pragma unroll
        for (int b2 = 0; b2 < NB; ++b2) { hm += s_hm[b2]; st += s_st[b2]; }
        hm *= (1.0f / (float)NB);
        st *= (1.0f / (float)NB);
        const float paf = s_paf[0] * (1.0f / (float)PAF_ELEMS);
        out[0] = hm + paf + st;
    }
}

extern "C" void kernel_launch(void* const* d_in, const int* in_sizes, int n_in,
                              void* d_out, int out_size, void* d_ws, size_t ws_size,
                              hipStream_t stream) {
    const float* hm_pred  = (const float*)d_in[0];
    const float* paf_pred = (const float*)d_in[1];
    const float* hm_gt    = (const float*)d_in[2];
    const float* paf_gt   = (const float*)d_in[3];
    const int*   coords   = (const int*)d_in[4];

    // workspace layout (all 16B-aligned): 544 f32 | 544 i32 | 1024 f32
    float* per_kp   = (float*)d_ws;
    int*   amax     = (int*)((char*)d_ws + BK * sizeof(float));
    float* partials = (float*)((char*)d_ws + 2 * BK * sizeof(float));

    heatmap_kernel<<<BK, 256, 0, stream>>>(hm_pred, hm_gt, per_kp, amax);
    paf_kernel<<<PAF_BLOCKS, 256, 0, stream>>>(paf_pred, paf_gt, partials);
    finalize_kernel<<<1, 256, 0, stream>>>(per_kp, amax, partials, coords,
                                           (float*)d_out);
}